// Transformer_F_74071005987392
// MI455X (gfx1250) — compile-verified
//
#include <hip/hip_runtime.h>

// Transformer_F: Z <- Z + (1/h) * sum_j ((Z Q_j Z^T)(Z V_j^T))/n , 4 layers.
// Reassociated:  Z <- Z (I + W_b),  W_b = (1/(n h)) sum_j Q_j (Z^T Z) V_j^T
// We store Wt = W^T so every fragment everywhere is a contiguous 64-bit load:
//   G = Z^T Z              (symmetric; Z slab staged K-pair-interleaved in LDS)
//   U = G * Q^T  (= T^T)   (A from G rows, B from Q rows -> global_load_b64)
//   Wt += s * V * U        (A from V rows, B from pair-interleaved U in LDS)
//   Z <- Z + Z * W         (A from row-major Z slab, B from Wt rows)
// All GEMMs via V_WMMA_F32_16X16X4_F32 (fp32, matches reference precision).

typedef float v2f __attribute__((ext_vector_type(2)));
typedef float v8f __attribute__((ext_vector_type(8)));

#define BB      8
#define NN      2048
#define DD      128
#define NLAYER  4
#define NHEAD   8

#define GSPLIT  32
#define GCHUNK  (NN / GSPLIT)   // 64 rows of Z per G-workgroup
#define UROWS   128             // rows of Z per update-workgroup

__device__ __forceinline__ v8f vzero8() {
    v8f z;
#pragma unroll
    for (int i = 0; i < 8; ++i) z[i] = 0.0f;
    return z;
}

// D(16x16,f32) = A(16x4,f32) x B(4x16,f32) + C
__device__ __forceinline__ v8f wmma4(v2f a, v2f b, v8f c) {
    return __builtin_amdgcn_wmma_f32_16x16x4_f32(
        /*neg_a=*/false, a, /*neg_b=*/false, b,
        /*c_mod=*/(short)0, c, /*reuse_a=*/false, /*reuse_b=*/false);
}

__device__ __forceinline__ v2f ld2(const float* p) {        // aligned 8B load
    return *(const v2f*)p;
}

__global__ void zero_kernel(float* __restrict__ p, int n) {
    int i = blockIdx.x * blockDim.x + threadIdx.x;
    if (i < n) p[i] = 0.0f;
}

// ---------------------------------------------------------------------------
// G[b] += Z_chunk^T * Z_chunk  (partial Gram, atomically accumulated)
// grid: (GSPLIT, B), block 256 (8 waves). Wave w owns G rows [16w,16w+16).
// Z slab staged pair-interleaved: Z[m][d] -> sZ[(m/2)*2D + 2d + (m&1)]
// so A and B fragments are single aligned ds_load_b64.
// ---------------------------------------------------------------------------
__global__ __launch_bounds__(256) void g_kernel(const float* __restrict__ Z,
                                                float* __restrict__ G) {
    __shared__ float sZ[GCHUNK * DD];                  // 32 KB
    const int b    = blockIdx.y;
    const int spl  = blockIdx.x;
    const int tid  = threadIdx.x;
    const int wave = tid >> 5;
    const int lane = tid & 31;
    const int half = lane >> 4;                        // 0: K 0/1, 1: K 2/3
    const int l16  = lane & 15;

    const float4* Zb4 = (const float4*)(Z + ((size_t)b * NN + (size_t)spl * GCHUNK) * DD);
    for (int i = tid; i < GCHUNK * DD / 4; i += 256) {
        float4 v = Zb4[i];
        const int m = (4 * i) / DD, d = (4 * i) % DD;
        float* dst = &sZ[(m >> 1) * (2 * DD) + 2 * d + (m & 1)];
        dst[0] = v.x; dst[2] = v.y; dst[4] = v.z; dst[6] = v.w;
    }
    __syncthreads();

    v8f acc[8];
#pragma unroll
    for (int t = 0; t < 8; ++t) acc[t] = vzero8();

    const int arow = 16 * wave + l16;                  // col of Z == row of Z^T
    for (int k = 0; k < GCHUNK; k += 4) {
        const int k0 = k + 2 * half;                   // even
        const float* base = &sZ[(k0 >> 1) * (2 * DD)];
        v2f a = ld2(base + 2 * arow);                  // {Z[k0][arow], Z[k0+1][arow]}
#pragma unroll
        for (int tc = 0; tc < 8; ++tc) {
            v2f bv = ld2(base + 2 * (16 * tc + l16));
            acc[tc] = wmma4(a, bv, acc[tc]);
        }
    }

    float* Gb = G + (size_t)b * DD * DD;
#pragma unroll
    for (int tc = 0; tc < 8; ++tc)
#pragma unroll
        for (int r = 0; r < 8; ++r) {
            const int row = 16 * wave + r + 8 * half;  // C layout: M split by lane half
            const int col = 16 * tc + l16;
            atomicAdd(&Gb[row * DD + col], acc[tc][r]);
        }
}

// ---------------------------------------------------------------------------
// Wt[b] += scale * V_j * (G[b] * Q_j^T)      grid: (NHEAD, B), block 256
// (G symmetric => G*Q^T = T^T;  V*(T^T) = P^T; accumulating W transposed)
// ---------------------------------------------------------------------------
__global__ __launch_bounds__(256) void w_kernel(const float* __restrict__ G,
                                                const float* __restrict__ Qm,
                                                const float* __restrict__ Vm,
                                                float* __restrict__ Wt) {
    __shared__ float sU[DD * DD];                      // 64 KB, pair-interleaved U
    const int b    = blockIdx.y;
    const int head = blockIdx.x;
    const int tid  = threadIdx.x;
    const int wave = tid >> 5;
    const int lane = tid & 31;
    const int half = lane >> 4;
    const int l16  = lane & 15;

    const float* Gb = G  + (size_t)b    * DD * DD;
    const float* Q  = Qm + (size_t)head * DD * DD;     // [D,D] row-major, L2-resident
    const float* V  = Vm + (size_t)head * DD * DD;

    const int arow = 16 * wave + l16;

    // ---- stage 1: U = G * Q^T  (all fragments contiguous global b64) ----
    v8f acc[8];
#pragma unroll
    for (int t = 0; t < 8; ++t) acc[t] = vzero8();
    for (int k = 0; k < DD; k += 4) {
        const int k0 = k + 2 * half;
        v2f a = ld2(Gb + arow * DD + k0);              // G row (symmetric => Z^T side free)
#pragma unroll
        for (int tc = 0; tc < 8; ++tc) {
            v2f bv = ld2(Q + (16 * tc + l16) * DD + k0);   // Q^T col = Q row
            acc[tc] = wmma4(a, bv, acc[tc]);
        }
    }
    // write U pair-interleaved: U[row][col] -> sU[(row/2)*2D + 2col + (row&1)]
#pragma unroll
    for (int tc = 0; tc < 8; ++tc)
#pragma unroll
        for (int r = 0; r < 8; ++r) {
            const int row = 16 * wave + r + 8 * half;
            const int col = 16 * tc + l16;
            sU[(row >> 1) * (2 * DD) + 2 * col + (row & 1)] = acc[tc][r];
        }
    __syncthreads();

    // ---- stage 2: Pt = V * U ----
#pragma unroll
    for (int t = 0; t < 8; ++t) acc[t] = vzero8();
    for (int k = 0; k < DD; k += 4) {
        const int k0 = k + 2 * half;
        v2f a = ld2(V + arow * DD + k0);               // V row
        const float* base = &sU[(k0 >> 1) * (2 * DD)];
#pragma unroll
        for (int tc = 0; tc < 8; ++tc) {
            v2f bv = ld2(base + 2 * (16 * tc + l16));  // U pairs along k
            acc[tc] = wmma4(a, bv, acc[tc]);
        }
    }

    const float scale = 1.0f / ((float)NN * (float)NHEAD);
    float* Wb = Wt + (size_t)b * DD * DD;
#pragma unroll
    for (int tc = 0; tc < 8; ++tc)
#pragma unroll
        for (int r = 0; r < 8; ++r) {
            const int row = 16 * wave + r + 8 * half;
            const int col = 16 * tc + l16;
            atomicAdd(&Wb[row * DD + col], scale * acc[tc][r]);
        }
}

// ---------------------------------------------------------------------------
// Zout rows = Zin rows + Zin rows * W   (W consumed via Wt rows, row-local
// => in-place safe).   grid: (N/UROWS, B), block 256
// ---------------------------------------------------------------------------
__global__ __launch_bounds__(256) void update_kernel(const float* __restrict__ Zin,
                                                     const float* __restrict__ Wt,
                                                     float* __restrict__ Zout) {
    __shared__ float lds[UROWS * DD];                  // 64 KB row-major Z slab
    const int b    = blockIdx.y;
    const int rb   = blockIdx.x;
    const int tid  = threadIdx.x;
    const int wave = tid >> 5;
    const int lane = tid & 31;
    const int half = lane >> 4;
    const int l16  = lane & 15;

    const float4* Zb4 = (const float4*)(Zin + ((size_t)b * NN + (size_t)rb * UROWS) * DD);
    float4* l4 = (float4*)lds;
    for (int i = tid; i < UROWS * DD / 4; i += 256) l4[i] = Zb4[i];
    __syncthreads();

    v8f acc[8];
#pragma unroll
    for (int t = 0; t < 8; ++t) acc[t] = vzero8();

    const float* Wb = Wt + (size_t)b * DD * DD;        // 64 KB, L2-resident
    const int arow = 16 * wave + l16;
    for (int k = 0; k < DD; k += 4) {
        const int k0 = k + 2 * half;
        v2f a = ld2(lds + arow * DD + k0);             // Z row pairs (contiguous)
#pragma unroll
        for (int tc = 0; tc < 8; ++tc) {
            v2f bv = ld2(Wb + (16 * tc + l16) * DD + k0);  // W[k][n] = Wt[n][k] pairs
            acc[tc] = wmma4(a, bv, acc[tc]);
        }
    }

    float* Ob = Zout + ((size_t)b * NN + (size_t)rb * UROWS) * DD;
#pragma unroll
    for (int tc = 0; tc < 8; ++tc)
#pragma unroll
        for (int r = 0; r < 8; ++r) {
            const int row = 16 * wave + r + 8 * half;
            const int col = 16 * tc + l16;
            Ob[row * DD + col] = lds[row * DD + col] + acc[tc][r];  // residual add
        }
}

// ---------------------------------------------------------------------------
extern "C" void kernel_launch(void* const* d_in, const int* in_sizes, int n_in,
                              void* d_out, int out_size, void* d_ws, size_t ws_size,
                              hipStream_t stream) {
    (void)in_sizes; (void)n_in; (void)out_size; (void)ws_size;

    const float* Z0 = (const float*)d_in[0];           // [B,N,D] f32
    const float* Vm = (const float*)d_in[1];           // [L,H,1,D,D] f32
    const float* Qm = (const float*)d_in[2];           // [L,H,1,D,D] f32
    float* Zout = (float*)d_out;                       // [B,N,D] f32

    float* G  = (float*)d_ws;                          // [B,D,D]  512 KB
    float* Wt = G + (size_t)BB * DD * DD;              // [B,D,D]  512 KB (W transposed)

    const int nzero = 2 * BB * DD * DD;                // zero G and Wt together

    for (int layer = 0; layer < NLAYER; ++layer) {
        const float* Zin = (layer == 0) ? Z0 : (const float*)Zout;
        const size_t poff = (size_t)layer * NHEAD * DD * DD;

        zero_kernel<<<(nzero + 255) / 256, 256, 0, stream>>>(G, nzero);
        g_kernel<<<dim3(GSPLIT, BB), 256, 0, stream>>>(Zin, G);
        w_kernel<<<dim3(NHEAD, BB), 256, 0, stream>>>(G, Qm + poff, Vm + poff, Wt);
        update_kernel<<<dim3(NN / UROWS, BB), 256, 0, stream>>>(Zin, Wt, Zout);
    }
}